// GKernel_66048007078604
// MI455X (gfx1250) — compile-verified
//
#include <hip/hip_runtime.h>
#include <hip/hip_bf16.h>

typedef float v2f __attribute__((ext_vector_type(2)));
typedef float v8f __attribute__((ext_vector_type(8)));

#define NUM_H 12
#define KVOL 125          // 5*5*5 spatial points
#define BSTRIDE 128       // LDS row stride for sampling matrix (125 rows x 128 cols)
#define MSPLIT 4          // split M=1024 across 4 blocks -> 576 blocks total

// Blocks: (o,i) pair x MSPLIT. 256 threads = 8 waves (wave32).
// Computes out[c_out, o, c_in, i, p] = sum_q weight[c_out, c_in, h(o,i), q] * S_o[q, p]
__global__ __launch_bounds__(256)
void gconv_weight_transform(const float* __restrict__ in_H,
                            const float* __restrict__ out_H,
                            const float* __restrict__ grid_H,
                            const float* __restrict__ grid_Rn,
                            const float* __restrict__ weight,
                            float* __restrict__ d_out)
{
    __shared__ float sB[KVOL * BSTRIDE];   // 64000 B: S_o stored as [q][p]
    __shared__ int   s_h;

    const int tid  = threadIdx.x;
    const int lane = tid & 31;
    const int wv   = tid >> 5;        // wave id 0..7
    const int lo   = lane & 15;       // lane within half-wave
    const int hi   = lane >> 4;       // half-wave select

    const int pair  = blockIdx.x >> 2;       // (o,i) pair index 0..143
    const int split = blockIdx.x & 3;        // M-split 0..3
    const int o = pair / NUM_H;
    const int i = pair % NUM_H;

    // ---- Phase A: zero the sampling matrix ----
    for (int t = tid; t < KVOL * BSTRIDE; t += 256) sB[t] = 0.0f;

    // ---- Phase B: nearest group element h = argmin_h ||out_H[o]^T in_H[i] - grid_H[h]||^2 ----
    if (tid == 0) {
        float P[9];
        for (int a = 0; a < 3; ++a)
            for (int b = 0; b < 3; ++b) {
                float s = 0.0f;
                for (int c = 0; c < 3; ++c)
                    s += out_H[o * 9 + c * 3 + a] * in_H[i * 9 + c * 3 + b];
                P[a * 3 + b] = s;
            }
        int best = 0; float bestd = 3.4e38f;
        for (int h = 0; h < NUM_H; ++h) {
            float d = 0.0f;
            for (int e = 0; e < 9; ++e) {
                float t = P[e] - grid_H[h * 9 + e];
                d += t * t;
            }
            if (d < bestd) { bestd = d; best = h; }
        }
        s_h = best;
    }

    __syncthreads();   // sB zero complete before scatter

    // ---- Phase C: build trilinear sampling matrix S_o[q][p] (thread p owns column p) ----
    if (tid < KVOL) {
        const int p = tid;
        const float r0 = grid_Rn[p * 3 + 0];
        const float r1 = grid_Rn[p * 3 + 1];
        const float r2 = grid_Rn[p * 3 + 2];
        // g = out_H[o]^T * r  (g[a] = sum_j out_H[j][a] * r[j])
        const float g0 = out_H[o*9+0]*r0 + out_H[o*9+3]*r1 + out_H[o*9+6]*r2;
        const float g1 = out_H[o*9+1]*r0 + out_H[o*9+4]*r1 + out_H[o*9+7]*r2;
        const float g2 = out_H[o*9+2]*r0 + out_H[o*9+5]*r1 + out_H[o*9+8]*r2;
        const float ix = (g0 + 1.0f) * 2.0f;   // (g+1)*0.5*(5-1)
        const float iy = (g1 + 1.0f) * 2.0f;
        const float iz = (g2 + 1.0f) * 2.0f;
        const float fx = floorf(ix), fy = floorf(iy), fz = floorf(iz);
        const float tx = ix - fx, ty = iy - fy, tz = iz - fz;
        #pragma unroll
        for (int dz = 0; dz < 2; ++dz)
        #pragma unroll
        for (int dy = 0; dy < 2; ++dy)
        #pragma unroll
        for (int dx = 0; dx < 2; ++dx) {
            const float cx = fx + dx, cy = fy + dy, cz = fz + dz;
            float w = (dx ? tx : 1.0f - tx) * (dy ? ty : 1.0f - ty) * (dz ? tz : 1.0f - tz);
            const bool inb = (cx >= 0.f) & (cx <= 4.f) & (cy >= 0.f) & (cy <= 4.f)
                           & (cz >= 0.f) & (cz <= 4.f);
            const int cxi = min(max((int)cx, 0), 4);
            const int cyi = min(max((int)cy, 0), 4);
            const int czi = min(max((int)cz, 0), 4);
            const int q = (czi * 5 + cyi) * 5 + cxi;
            sB[q * BSTRIDE + p] += inb ? w : 0.0f;   // column p owned by this thread
        }
    }

    __syncthreads();

    const int h = s_h;

    // ---- Phase D: GEMM  D[256 x 125] = W[256 x 125] * S_o[125 x 125] via WMMA f32 16x16x4 ----
    // This block owns M rows [split*256, split*256+256); wave wv owns 2 M-tiles.
    for (int mt = 0; mt < 2; ++mt) {
        // A-fragment source row for this lane (ISA 16x4 f32 A layout: lanes 0-15 -> M rows,
        // VGPR0/1 hold K, K+1; half-wave hi adds K-offset 2).
        const int row = split * 256 + wv * 32 + mt * 16 + lo;
        const float* Wp = weight +
            (size_t)((((row & 31) * 32 + (row >> 5)) * NUM_H + h)) * KVOL;

        v8f acc[8];
        #pragma unroll
        for (int n = 0; n < 8; ++n) acc[n] = (v8f){0,0,0,0,0,0,0,0};

        // 31 full k-steps covering q = 0..123
        for (int k = 0; k < 31; ++k) {
            const int kk = 4 * k + 2 * hi;
            v2f a;
            a.x = Wp[kk];
            a.y = Wp[kk + 1];
            #pragma unroll
            for (int n = 0; n < 8; ++n) {
                v2f b;
                b.x = sB[kk * BSTRIDE + n * 16 + lo];
                b.y = sB[(kk + 1) * BSTRIDE + n * 16 + lo];
                acc[n] = __builtin_amdgcn_wmma_f32_16x16x4_f32(
                    /*neg_a=*/false, a, /*neg_b=*/false, b,
                    /*c_mod=*/(short)0, acc[n],
                    /*reuse_a=*/false, /*reuse_b=*/false);
            }
        }

        // Tail rank-1 update for q = 124 (keeps LDS un-padded and loads in-bounds).
        #pragma unroll
        for (int v = 0; v < 8; ++v) {
            const int rowv = split * 256 + wv * 32 + mt * 16 + v + 8 * hi;
            const float av = weight[
                (size_t)((((rowv & 31) * 32 + (rowv >> 5)) * NUM_H + h)) * KVOL + 124];
            #pragma unroll
            for (int n = 0; n < 8; ++n)
                acc[n][v] += av * sB[124 * BSTRIDE + n * 16 + lo];
        }

        // Store: D element (M = v + 8*hi, N = lo) per C/D layout; output layout
        // (c_out, o, c_in, i, p) with GEMM row = c_in*32 + c_out. Output is
        // write-once, never re-read -> non-temporal stores keep L2 for weights.
        #pragma unroll
        for (int n = 0; n < 8; ++n) {
            const int p = n * 16 + lo;
            if (p < KVOL) {
                #pragma unroll
                for (int v = 0; v < 8; ++v) {
                    const int rowv = split * 256 + wv * 32 + mt * 16 + v + 8 * hi;
                    const int c_in = rowv >> 5;
                    const int c_out = rowv & 31;
                    const size_t oidx =
                        ((((size_t)c_out * NUM_H + o) * 32 + c_in) * NUM_H + i) * KVOL + p;
                    __builtin_nontemporal_store(acc[n][v], d_out + oidx);
                }
            }
        }
    }
}

extern "C" void kernel_launch(void* const* d_in, const int* in_sizes, int n_in,
                              void* d_out, int out_size, void* d_ws, size_t ws_size,
                              hipStream_t stream) {
    (void)in_sizes; (void)n_in; (void)d_ws; (void)ws_size; (void)out_size;
    const float* in_H    = (const float*)d_in[0];
    const float* out_H   = (const float*)d_in[1];
    const float* grid_H  = (const float*)d_in[2];
    const float* grid_Rn = (const float*)d_in[3];
    const float* weight  = (const float*)d_in[4];
    float* out = (float*)d_out;

    dim3 grid(NUM_H * NUM_H * MSPLIT);   // 576 blocks: 144 (o,i) pairs x 4 M-splits
    dim3 block(256);                     // 8 wave32 waves
    gconv_weight_transform<<<grid, block, 0, stream>>>(
        in_H, out_H, grid_H, grid_Rn, weight, out);
}